// NodeToNodeLayer_82162724372842
// MI455X (gfx1250) — compile-verified
//
#include <hip/hip_runtime.h>
#include <hip/hip_bf16.h>

typedef float v2f __attribute__((ext_vector_type(2)));
typedef float v8f __attribute__((ext_vector_type(8)));

#define D_FEAT 64
#define TWO_D  128   // concat(agg, x)
#define HID    128
#define OUTF   128
#define CS     132   // comb LDS row stride (floats) -> conflict-free A-frag reads
#define WFRAG_FLOATS (HID * TWO_D)   // 16384 floats = 64 KB, fragment-ordered

// ---------------- kernel 1: zero the scatter accumulators ----------------
__global__ void zero_ws_kernel(float* __restrict__ p, long long n) {
  long long i = (long long)blockIdx.x * blockDim.x + threadIdx.x;
  long long stride = (long long)gridDim.x * blockDim.x;
  for (; i < n; i += stride) p[i] = 0.0f;
}

// ---------------- kernel 2: wave-per-edge scatter-add (mean numerator + counts) ----
__global__ __launch_bounds__(256)
void scatter_mean_kernel(const float* __restrict__ nf,
                         const long long* __restrict__ ei,  // [2, E] int64
                         float* __restrict__ agg,
                         float* __restrict__ cnt,
                         int nEdges) {
  int warp = (blockIdx.x * blockDim.x + threadIdx.x) >> 5;
  int lane = threadIdx.x & 31;
  if (warp >= nEdges) return;
  int src = (int)ei[warp];                       // senders = row 0
  int tgt = (int)ei[(long long)nEdges + warp];   // receivers = row 1
  float f0 = nf[(long long)src * D_FEAT + lane];
  float f1 = nf[(long long)src * D_FEAT + 32 + lane];
  __hip_atomic_fetch_add(&agg[(long long)tgt * D_FEAT + lane], f0,
                         __ATOMIC_RELAXED, __HIP_MEMORY_SCOPE_AGENT);
  __hip_atomic_fetch_add(&agg[(long long)tgt * D_FEAT + 32 + lane], f1,
                         __ATOMIC_RELAXED, __HIP_MEMORY_SCOPE_AGENT);
  if (lane == 0)
    __hip_atomic_fetch_add(&cnt[tgt], 1.0f,
                           __ATOMIC_RELAXED, __HIP_MEMORY_SCOPE_AGENT);
}

// Stage weight matrix W[n][k] (row-major, 128x128) into WMMA B-fragment order:
//   B[k][n] = W[n][k];  lane l, VGPR j of frag(ks,nt) holds B[4ks+j+2*(l>>4)][16nt+(l&15)]
//   -> wfrag[ ((ks*8+nt)*32 + lane)*2 + j ]
__device__ __forceinline__ void stage_wfrag(const float* __restrict__ W,
                                            float* __restrict__ wfrag, int tid) {
  for (int i = tid; i < HID * TWO_D; i += 256) {
    int n = i >> 7, k = i & 127;
    int ks = k >> 2, j = k & 1, hf = (k >> 1) & 1;
    int idx = (((ks * 8 + (n >> 4)) * 32) + (hf * 16 + (n & 15))) * 2 + j;
    wfrag[idx] = W[i];
  }
}

// ---------------- kernel 3: fused mean-normalize + MLP via fp32 WMMA ----------------
// block = 256 threads (8 waves), 128 nodes per block, 16-node M-tile per wave.
__global__ __launch_bounds__(256)
void gnn_mlp_kernel(const float* __restrict__ nf,
                    const float* __restrict__ agg,
                    const float* __restrict__ cnt,
                    const float* __restrict__ W1, const float* __restrict__ b1,
                    const float* __restrict__ W2, const float* __restrict__ b2,
                    float* __restrict__ out, int nNodes) {
  extern __shared__ float smem[];
  float* comb  = smem;                      // [128][CS], combined -> reused for h
  float* wfrag = smem + 128 * CS;           // 16384 floats, fragment-ordered weights
  float* lbias = wfrag + WFRAG_FLOATS;      // [0..127]=b1, [128..255]=b2

  const int tid  = threadIdx.x;
  const int wave = tid >> 5;
  const int lane = tid & 31;
  const int row  = lane & 15;       // M (A) / N (B,C,D) index within 16-tile
  const int half = lane >> 4;       // 0: lanes 0-15, 1: lanes 16-31
  const int tileBase = blockIdx.x * 128;

  // ---- stage combined = [agg/cnt | x], biases, and W1 fragments into LDS
  for (int i = tid; i < 128 * TWO_D; i += 256) {
    int r = i >> 7, c = i & 127;
    int node = tileBase + r;
    float v = 0.0f;
    if (node < nNodes) {
      if (c < D_FEAT) {
        float cn = cnt[node];
        v = agg[(long long)node * D_FEAT + c] / fmaxf(cn, 1.0f);
      } else {
        v = nf[(long long)node * D_FEAT + (c - D_FEAT)];
      }
    }
    comb[r * CS + c] = v;
  }
  if (tid < 128) {
    lbias[tid]       = b1[tid];
    lbias[128 + tid] = b2[tid];
  }
  stage_wfrag(W1, wfrag, tid);
  __syncthreads();

  const v2f* wfp = (const v2f*)wfrag;
  // A-frag base: lane holds row (wave*16+row), K = 4*ks + 2*half + {0,1}
  const float* arow = &comb[(wave * 16 + row) * CS + 2 * half];

  // ================= GEMM1: h = relu(comb @ W1^T + b1) =================
  v2f afrag[32];
#pragma unroll
  for (int ks = 0; ks < 32; ++ks)
    afrag[ks] = *(const v2f*)(arow + 4 * ks);

  for (int nt = 0; nt < 8; ++nt) {
    float bias = lbias[nt * 16 + row];
    v8f acc;
#pragma unroll
    for (int j = 0; j < 8; ++j) acc[j] = bias;
    v2f bfrag[32];
#pragma unroll
    for (int ks = 0; ks < 32; ++ks)
      bfrag[ks] = wfp[(ks * 8 + nt) * 32 + lane];   // one ds_load_b64 each
#pragma unroll
    for (int ks = 0; ks < 32; ++ks)
      acc = __builtin_amdgcn_wmma_f32_16x16x4_f32(
          false, afrag[ks], false, bfrag[ks], (short)0, acc, false, false);
    // D layout: VGPR v, lane -> element [M = v + 8*half][N = row] of tile
#pragma unroll
    for (int v = 0; v < 8; ++v)
      comb[(wave * 16 + v + 8 * half) * CS + nt * 16 + row] = fmaxf(acc[v], 0.0f);
  }
  __syncthreads();

  // ---- swap W2 fragments into the same weight buffer
  stage_wfrag(W2, wfrag, tid);
  __syncthreads();

  // ================= GEMM2: out = h @ W2^T + b2 =================
#pragma unroll
  for (int ks = 0; ks < 32; ++ks)
    afrag[ks] = *(const v2f*)(arow + 4 * ks);

  for (int nt = 0; nt < 8; ++nt) {
    float bias = lbias[128 + nt * 16 + row];
    v8f acc;
#pragma unroll
    for (int j = 0; j < 8; ++j) acc[j] = bias;
    v2f bfrag[32];
#pragma unroll
    for (int ks = 0; ks < 32; ++ks)
      bfrag[ks] = wfp[(ks * 8 + nt) * 32 + lane];
#pragma unroll
    for (int ks = 0; ks < 32; ++ks)
      acc = __builtin_amdgcn_wmma_f32_16x16x4_f32(
          false, afrag[ks], false, bfrag[ks], (short)0, acc, false, false);
#pragma unroll
    for (int v = 0; v < 8; ++v) {
      int node = tileBase + wave * 16 + v + 8 * half;
      if (node < nNodes)
        out[(long long)node * OUTF + nt * 16 + row] = acc[v];
    }
  }
}

// ---------------- host launcher ----------------
extern "C" void kernel_launch(void* const* d_in, const int* in_sizes, int n_in,
                              void* d_out, int out_size, void* d_ws, size_t ws_size,
                              hipStream_t stream) {
  const float*     nf = (const float*)d_in[0];
  const long long* ei = (const long long*)d_in[1];  // int64 per reference
  const float*     W1 = (const float*)d_in[2];
  const float*     b1 = (const float*)d_in[3];
  const float*     W2 = (const float*)d_in[4];
  const float*     b2 = (const float*)d_in[5];
  float* out = (float*)d_out;

  const int N = in_sizes[0] / D_FEAT;
  const int E = in_sizes[1] / 2;

  float* agg = (float*)d_ws;                 // [N, 64]
  float* cnt = agg + (size_t)N * D_FEAT;     // [N]

  // 1) zero accumulators
  long long zn = (long long)N * (D_FEAT + 1);
  zero_ws_kernel<<<2048, 256, 0, stream>>>(agg, zn);

  // 2) edge scatter-add (one wave32 per edge)
  int eBlocks = (E + 7) / 8;                 // 8 waves per 256-thread block
  scatter_mean_kernel<<<eBlocks, 256, 0, stream>>>(nf, ei, agg, cnt, E);

  // 3) fused normalize + 2-layer MLP with fp32 WMMA
  int nBlocks = (N + 127) / 128;
  size_t smemBytes = (size_t)(128 * CS + WFRAG_FLOATS + 256) * sizeof(float); // ~131 KB
  gnn_mlp_kernel<<<nBlocks, 256, smemBytes, stream>>>(nf, agg, cnt,
                                                      W1, b1, W2, b2, out, N);
}